// RedConv_88656714924912
// MI455X (gfx1250) — compile-verified
//
#include <hip/hip_runtime.h>
#include <hip/hip_bf16.h>
#include <math.h>

#define DD 128
#define NEG_SLOPE 0.2f
#define LAMBDA 0.1f
#define SXS 129   // LDS row stride (floats) to avoid bank conflicts

typedef float v2f __attribute__((ext_vector_type(2)));
typedef float v8f __attribute__((ext_vector_type(8)));

// ---- helpers ------------------------------------------------------------

__device__ __forceinline__ unsigned fenc(float f) {
    unsigned u = __float_as_uint(f);
    return (u & 0x80000000u) ? ~u : (u | 0x80000000u);
}
__device__ __forceinline__ float fdec(unsigned u) {
    u = (u & 0x80000000u) ? (u & 0x7FFFFFFFu) : ~u;
    return __uint_as_float(u);
}
__device__ __forceinline__ float wave_sum(float v) {
    for (int o = 16; o > 0; o >>= 1) v += __shfl_xor(v, o, 32);
    return v;
}
__device__ __forceinline__ float leaky(float v) {
    return v >= 0.0f ? v : NEG_SLOPE * v;
}

// ---- init ---------------------------------------------------------------

__global__ void k_init_nodes(float* deg, float* zk, float* zq,
                             unsigned* mk, unsigned* mq,
                             float* ssum, float* aggdot, int n) {
    int i = blockIdx.x * blockDim.x + threadIdx.x;
    if (i >= n) return;
    deg[i] = 0.f; zk[i] = 0.f; zq[i] = 0.f; ssum[i] = 0.f; aggdot[i] = 0.f;
    mk[i] = 0x007FFFFFu;  // fenc(-inf)
    mq[i] = 0x007FFFFFu;
}

// ---- degree / norm ------------------------------------------------------

__global__ void k_deg(const int* __restrict__ col, float* deg, int etot) {
    int e = blockIdx.x * blockDim.x + threadIdx.x;
    if (e < etot) atomicAdd(&deg[col[e]], 1.0f);
}

__global__ void k_dinv(float* deg, int n) {
    int i = blockIdx.x * blockDim.x + threadIdx.x;
    if (i >= n) return;
    float d = deg[i];
    deg[i] = d > 0.0f ? rsqrtf(d) : 0.0f;
}

// ---- GEMM: xw = x @ W_gcn  via V_WMMA_F32_16X16X4_F32 -------------------
// block = 256 threads (8 waves). One 16-row M tile per block; wave w owns
// the 16-column N tile [16w, 16w+16). K loop of 32 fp32 WMMAs.

__global__ void __launch_bounds__(256)
k_gemm_wmma(const float* __restrict__ x, const float* __restrict__ W,
            float* __restrict__ xw, int nrows) {
    __shared__ float sx[16 * SXS];
    const int m0 = blockIdx.x * 16;
    // cooperative load of 16x128 x-tile into LDS (clamped for tail)
    for (int i = threadIdx.x; i < 16 * DD; i += 256) {
        int r = i >> 7, c = i & (DD - 1);
        int rr = m0 + r; if (rr >= nrows) rr = nrows - 1;
        sx[r * SXS + c] = x[(size_t)rr * DD + c];
    }
    __syncthreads();

    const int wave = threadIdx.x >> 5;
    const int lane = threadIdx.x & 31;
    const int hl   = lane >> 4;     // half-wave select
    const int l    = lane & 15;
    const int n0   = wave * 16;

    v8f c = {};
    for (int k = 0; k < DD; k += 4) {
        const int ka = k + 2 * hl;
        v2f a, b;
        // A 16x4 fp32 layout: lane = hl*16 + l holds (M=l, K=ka / ka+1)
        a.x = sx[l * SXS + ka];
        a.y = sx[l * SXS + ka + 1];
        // B 4x16 fp32 layout (transposed analog): lane holds (K=ka / ka+1, N=l)
        b.x = W[(size_t)ka * DD + n0 + l];
        b.y = W[(size_t)(ka + 1) * DD + n0 + l];
#if defined(__HIP_DEVICE_COMPILE__)
        c = __builtin_amdgcn_wmma_f32_16x16x4_f32(
                false, a, false, b, (short)0, c, false, false);
#else
        (void)a; (void)b;
#endif
    }
    // C/D layout: VGPR v -> (M = v + 8*hl, N = l)
    for (int v = 0; v < 8; ++v) {
        int row = m0 + hl * 8 + v;
        if (row < nrows) xw[(size_t)row * DD + n0 + l] = c[v];
    }
}

// ---- x_transform[col] += norm * xw[row]  (one wave per edge) ------------

__global__ void __launch_bounds__(256)
k_scatter_transform(const int* __restrict__ row, const int* __restrict__ col,
                    const float* __restrict__ xw, const float* __restrict__ dinv,
                    float* xt, int etot) {
    int e = blockIdx.x * 8 + (threadIdx.x >> 5);
    if (e >= etot) return;
    int lane = threadIdx.x & 31;
    int r = row[e], c = col[e];
    float norm = dinv[r] * dinv[c];
    float4 v = *(const float4*)(xw + (size_t)r * DD + lane * 4);
    float* dst = xt + (size_t)c * DD + lane * 4;
    atomicAdd(dst + 0, norm * v.x);
    atomicAdd(dst + 1, norm * v.y);
    atomicAdd(dst + 2, norm * v.z);
    atomicAdd(dst + 3, norm * v.w);
}

// ---- per-node: bias add, key/query scores, xrel/xroot dots --------------

__global__ void __launch_bounds__(256)
k_node_prep(float* __restrict__ xt, const float* __restrict__ x,
            const float* __restrict__ bg,
            const float* __restrict__ Wk, const float* __restrict__ bk,
            const float* __restrict__ Wq, const float* __restrict__ bq,
            const float* __restrict__ Wrel, const float* __restrict__ Wroot,
            float* ks, float* qs, float* xrel, float* xroot, int n) {
    int node = blockIdx.x * 8 + (threadIdx.x >> 5);
    if (node >= n) return;
    int lane = threadIdx.x & 31;
    size_t base = (size_t)node * DD + lane * 4;

    float4 xtv = *(float4*)(xt + base);
    float4 bgv = *(const float4*)(bg + lane * 4);
    xtv.x += bgv.x; xtv.y += bgv.y; xtv.z += bgv.z; xtv.w += bgv.w;
    *(float4*)(xt + base) = xtv;

    float4 wkv = *(const float4*)(Wk + lane * 4);
    float4 wqv = *(const float4*)(Wq + lane * 4);
    float kd = xtv.x * wkv.x + xtv.y * wkv.y + xtv.z * wkv.z + xtv.w * wkv.w;
    float qd = xtv.x * wqv.x + xtv.y * wqv.y + xtv.z * wqv.z + xtv.w * wqv.w;

    float4 xv  = *(const float4*)(x + base);
    float4 wrv = *(const float4*)(Wrel + lane * 4);
    float4 wov = *(const float4*)(Wroot + lane * 4);
    float rd  = xv.x * wrv.x + xv.y * wrv.y + xv.z * wrv.z + xv.w * wrv.w;
    float rtd = xv.x * wov.x + xv.y * wov.y + xv.z * wov.z + xv.w * wov.w;

    kd = wave_sum(kd); qd = wave_sum(qd); rd = wave_sum(rd); rtd = wave_sum(rtd);
    if (lane == 0) {
        ks[node] = leaky(kd + bk[0]);
        qs[node] = leaky(qd + bq[0]);
        xrel[node] = rd;
        xroot[node] = rtd;
    }
}

// ---- segment softmax pass 1: max (and fused aggdot scatter) -------------

__global__ void k_seg_max(const int* __restrict__ row, const int* __restrict__ col,
                          const float* __restrict__ ks, const float* __restrict__ qs,
                          const float* __restrict__ xrel,
                          unsigned* mk, unsigned* mq, float* aggdot, int etot) {
    int e = blockIdx.x * blockDim.x + threadIdx.x;
    if (e >= etot) return;
    int r = row[e], c = col[e];
    atomicMax(&mk[r], fenc(ks[c]));
    atomicMax(&mq[r], fenc(qs[c]));
    atomicAdd(&aggdot[c], xrel[r]);
}

// ---- segment softmax pass 2: denominator --------------------------------

__global__ void k_seg_sum(const int* __restrict__ row, const int* __restrict__ col,
                          const float* __restrict__ ks, const float* __restrict__ qs,
                          const unsigned* __restrict__ mk, const unsigned* __restrict__ mq,
                          float* zk, float* zq, int etot) {
    int e = blockIdx.x * blockDim.x + threadIdx.x;
    if (e >= etot) return;
    int r = row[e], c = col[e];
    atomicAdd(&zk[r], expf(ks[c] - fdec(mk[r])));
    atomicAdd(&zq[r], expf(qs[c] - fdec(mq[r])));
}

// ---- reweight scatter: rw[row] += w * x_transform[col] ------------------

__global__ void __launch_bounds__(256)
k_reweight(const int* __restrict__ row, const int* __restrict__ col,
           const float* __restrict__ xt,
           const float* __restrict__ ks, const float* __restrict__ qs,
           const unsigned* __restrict__ mk, const unsigned* __restrict__ mq,
           const float* __restrict__ zk, const float* __restrict__ zq,
           float* rwk, float* rwq, int etot) {
    int e = blockIdx.x * 8 + (threadIdx.x >> 5);
    if (e >= etot) return;
    int lane = threadIdx.x & 31;
    int r = row[e], c = col[e];
    float wk = expf(ks[c] - fdec(mk[r])) / zk[r];
    float wq = expf(qs[c] - fdec(mq[r])) / zq[r];
    float4 v = *(const float4*)(xt + (size_t)c * DD + lane * 4);
    float* pk = rwk + (size_t)r * DD + lane * 4;
    float* pq = rwq + (size_t)r * DD + lane * 4;
    atomicAdd(pk + 0, wk * v.x); atomicAdd(pk + 1, wk * v.y);
    atomicAdd(pk + 2, wk * v.z); atomicAdd(pk + 3, wk * v.w);
    atomicAdd(pq + 0, wq * v.x); atomicAdd(pq + 1, wq * v.y);
    atomicAdd(pq + 2, wq * v.z); atomicAdd(pq + 3, wq * v.w);
}

// ---- per-node key error -------------------------------------------------

__global__ void __launch_bounds__(256)
k_kerr(const float* __restrict__ rwk, const float* __restrict__ xt,
       float* kerr, int n) {
    int node = blockIdx.x * 8 + (threadIdx.x >> 5);
    if (node >= n) return;
    int lane = threadIdx.x & 31;
    size_t base = (size_t)node * DD + lane * 4;
    float4 a = *(const float4*)(rwk + base);
    float4 b = *(const float4*)(xt + base);
    float s = fabsf(a.x - b.x) + fabsf(a.y - b.y) +
              fabsf(a.z - b.z) + fabsf(a.w - b.w);
    s = wave_sum(s);
    if (lane == 0) kerr[node] = s;
}

// ---- per-edge score (non-self-loop edges only) --------------------------

__global__ void __launch_bounds__(256)
k_score(const int* __restrict__ row, const int* __restrict__ col,
        const float* __restrict__ rwq, const float* __restrict__ xt,
        const float* __restrict__ kerr, float* ssum, int e_real) {
    int e = blockIdx.x * 8 + (threadIdx.x >> 5);
    if (e >= e_real) return;
    int lane = threadIdx.x & 31;
    int r = row[e], c = col[e];
    float4 a = *(const float4*)(rwq + (size_t)r * DD + lane * 4);
    float4 b = *(const float4*)(xt  + (size_t)c * DD + lane * 4);
    float q = fabsf(a.x - b.x) + fabsf(a.y - b.y) +
              fabsf(a.z - b.z) + fabsf(a.w - b.w);
    q = wave_sum(q);
    if (lane == 0) atomicAdd(&ssum[r], kerr[r] - q);
}

// ---- final combine ------------------------------------------------------

__global__ void k_final(const float* __restrict__ aggdot,
                        const float* __restrict__ xroot,
                        const float* __restrict__ brel,
                        const float* __restrict__ ssum,
                        float* out, int n) {
    int i = blockIdx.x * blockDim.x + threadIdx.x;
    if (i >= n) return;
    float t = aggdot[i] + brel[0] + xroot[i];
    float fit = 1.0f / (1.0f + expf(-t));
    out[i] = fit - LAMBDA * ssum[i];
}

// ---- launch -------------------------------------------------------------

extern "C" void kernel_launch(void* const* d_in, const int* in_sizes, int n_in,
                              void* d_out, int out_size, void* d_ws, size_t ws_size,
                              hipStream_t stream) {
    const float* x     = (const float*)d_in[0];
    const int*   ei    = (const int*)d_in[1];
    const float* Wg    = (const float*)d_in[2];
    const float* bg    = (const float*)d_in[3];
    const float* Wk    = (const float*)d_in[4];
    const float* bk    = (const float*)d_in[5];
    const float* Wq    = (const float*)d_in[6];
    const float* bq    = (const float*)d_in[7];
    const float* Wrel  = (const float*)d_in[8];
    const float* brel  = (const float*)d_in[9];
    const float* Wroot = (const float*)d_in[10];
    float* out = (float*)d_out;

    const int N    = in_sizes[0] / DD;
    const int Etot = in_sizes[1] / 2;
    const int E    = Etot - N;          // non-self-loop edges
    const int* row = ei;                // edge_index[0] = source
    const int* col = ei + Etot;         // edge_index[1] = target

    // workspace carve-up (all offsets 256B aligned; N*DD*4 is already)
    char* ws = (char*)d_ws;
    size_t off = 0;
    auto carve = [&](size_t bytes) -> char* {
        char* p = ws + off;
        off += (bytes + 255) & ~(size_t)255;
        return p;
    };
    float*    xt     = (float*)carve((size_t)N * DD * 4);
    float*    rwk    = (float*)carve((size_t)N * DD * 4);
    float*    rwq    = (float*)carve((size_t)N * DD * 4);  // xt,rwk,rwq contiguous
    float*    xw     = (float*)carve((size_t)N * DD * 4);
    float*    deg    = (float*)carve((size_t)N * 4);       // becomes dinv in place
    float*    ks     = (float*)carve((size_t)N * 4);
    float*    qs     = (float*)carve((size_t)N * 4);
    unsigned* mk     = (unsigned*)carve((size_t)N * 4);
    unsigned* mq     = (unsigned*)carve((size_t)N * 4);
    float*    zk     = (float*)carve((size_t)N * 4);
    float*    zq     = (float*)carve((size_t)N * 4);
    float*    kerr   = (float*)carve((size_t)N * 4);
    float*    ssum   = (float*)carve((size_t)N * 4);
    float*    aggdot = (float*)carve((size_t)N * 4);
    float*    xrel   = (float*)carve((size_t)N * 4);
    float*    xroot  = (float*)carve((size_t)N * 4);
    (void)ws_size; (void)n_in; (void)out_size;

    const int nb_n  = (N + 255) / 256;        // thread-per-node
    const int nb_e  = (Etot + 255) / 256;     // thread-per-edge
    const int wb_n  = (N + 7) / 8;            // wave-per-node (256 thr = 8 waves)
    const int wb_e  = (Etot + 7) / 8;         // wave-per-edge
    const int wb_er = (E + 7) / 8;            // wave-per-real-edge
    const int nb_m  = (N + 15) / 16;          // gemm M tiles

    hipMemsetAsync(xt, 0, (size_t)3 * N * DD * 4, stream);  // xt, rwk, rwq
    k_init_nodes<<<nb_n, 256, 0, stream>>>(deg, zk, zq, mk, mq, ssum, aggdot, N);
    k_deg<<<nb_e, 256, 0, stream>>>(col, deg, Etot);
    k_dinv<<<nb_n, 256, 0, stream>>>(deg, N);
    k_gemm_wmma<<<nb_m, 256, 0, stream>>>(x, Wg, xw, N);
    k_scatter_transform<<<wb_e, 256, 0, stream>>>(row, col, xw, deg, xt, Etot);
    k_node_prep<<<wb_n, 256, 0, stream>>>(xt, x, bg, Wk, bk, Wq, bq, Wrel, Wroot,
                                          ks, qs, xrel, xroot, N);
    k_seg_max<<<nb_e, 256, 0, stream>>>(row, col, ks, qs, xrel, mk, mq, aggdot, Etot);
    k_seg_sum<<<nb_e, 256, 0, stream>>>(row, col, ks, qs, mk, mq, zk, zq, Etot);
    k_reweight<<<wb_e, 256, 0, stream>>>(row, col, xt, ks, qs, mk, mq, zk, zq,
                                         rwk, rwq, Etot);
    k_kerr<<<wb_n, 256, 0, stream>>>(rwk, xt, kerr, N);
    if (E > 0)
        k_score<<<wb_er, 256, 0, stream>>>(row, col, rwq, xt, kerr, ssum, E);
    k_final<<<nb_n, 256, 0, stream>>>(aggdot, xroot, brel, ssum, out, N);
}